// Conv_4973572129151
// MI455X (gfx1250) — compile-verified
//
#include <hip/hip_runtime.h>
#include <hip/hip_bf16.h>

#define MUL_S 128
#define MUL_V 64
#define D_IN  320            // MUL_S + 3*MUL_V
#define N_NODES 10000
#define N_EDGES 160000
#define DIM_KEY 64
#define NUM_FILTERS 32
#define W_NUMEL 384          // 128+128+64+64
#define MSG_W 768            // 128 + 384 + 192 + 64

#define INV_SQRT3   0.57735026918962576f
#define INV_SQRT128 0.088388347648318447f
#define INV_SQRT64  0.125f
#define OUT_SCALE   0.0072168783648703216f   // (1/sqrt(192)) * (1/10)

typedef float v2f __attribute__((ext_vector_type(2)));
typedef float v8f __attribute__((ext_vector_type(8)));

__device__ __forceinline__ v8f wmma4(v2f a, v2f b, v8f c) {
  // D = A(16x4,f32) * B(4x16,f32) + C(16x16,f32)
  return __builtin_amdgcn_wmma_f32_16x16x4_f32(false, a, false, b,
                                               (short)0, c, false, false);
}

__device__ __forceinline__ void atomicAddF(float* p, float v) {
  (void)__hip_atomic_fetch_add(p, v, __ATOMIC_RELAXED, __HIP_MEMORY_SCOPE_AGENT);
}

// ---------------------------------------------------------------- zero msg
__global__ void zero_kernel(float* __restrict__ p, long n) {
  long i = (long)blockIdx.x * blockDim.x + threadIdx.x;
  long stride = (long)gridDim.x * blockDim.x;
  for (; i < n; i += stride) p[i] = 0.0f;
}

// --------------------------------------------- kernel A: s_skip & x1 (WMMA)
__global__ void node_linear_kernel(const float* __restrict__ x,
                                   const float* __restrict__ W_si_s,
                                   const float* __restrict__ W_si_v,
                                   const float* __restrict__ W_l1_s,
                                   const float* __restrict__ W_l1_v,
                                   float* __restrict__ out_skip,   // d_out
                                   float* __restrict__ x1) {       // ws
  const int node0 = blockIdx.x * 16;
  const int wv   = threadIdx.x >> 5;
  const int lane = threadIdx.x & 31;
  const int hf   = lane >> 4;       // K half select
  const int lq   = lane & 15;       // M / N within half

  for (int task = wv; task < 40; task += 8) {
    v8f acc = {0.f,0.f,0.f,0.f,0.f,0.f,0.f,0.f};
    if (task < 16) {
      const int m  = task >> 3;     // 0 = skip, 1 = l1
      const int nt = task & 7;
      const float* W   = m ? W_l1_s : W_si_s;
      float*       dst = m ? x1 : out_skip;
      const float* arow = x + (long)(node0 + lq) * D_IN;
      #pragma unroll
      for (int k0 = 0; k0 < MUL_S; k0 += 4) {
        const int ka = k0 + 2 * hf;
        v2f a = *(const v2f*)(arow + ka);              // contiguous -> b64
        v2f b;
        b.x = W[(long)ka       * MUL_S + nt * 16 + lq];
        b.y = W[(long)(ka + 1) * MUL_S + nt * 16 + lq];
        acc = wmma4(a, b, acc);
      }
      #pragma unroll
      for (int r = 0; r < 8; ++r) {
        const int mrow = r + 8 * hf;
        dst[(long)(node0 + mrow) * D_IN + nt * 16 + lq] = acc[r] * INV_SQRT128;
      }
    } else {
      const int t2   = task - 16;       // 0..23
      const int m    = t2 / 12;
      const int rest = t2 % 12;
      const int ci   = rest >> 2;       // xyz component
      const int nt   = rest & 3;
      const float* W   = m ? W_l1_v : W_si_v;
      float*       dst = m ? x1 : out_skip;
      const float* arow = x + (long)(node0 + lq) * D_IN + MUL_S;
      #pragma unroll
      for (int k0 = 0; k0 < MUL_V; k0 += 4) {
        const int ka = k0 + 2 * hf;
        v2f a, b;
        a.x = arow[3 * ka + ci];
        a.y = arow[3 * (ka + 1) + ci];
        b.x = W[(long)ka       * MUL_V + nt * 16 + lq];
        b.y = W[(long)(ka + 1) * MUL_V + nt * 16 + lq];
        acc = wmma4(a, b, acc);
      }
      #pragma unroll
      for (int r = 0; r < 8; ++r) {
        const int mrow = r + 8 * hf;
        dst[(long)(node0 + mrow) * D_IN + MUL_S + 3 * (nt * 16 + lq) + ci] =
            acc[r] * INV_SQRT64;
      }
    }
  }
}

// ------------------------------------------------- kernel B: edges
// Block = 256 threads (8 waves). Processes TPB tiles of 16 edges.
// Phase 1: per-edge softmax attention weights p (wave per edge, 2 edges/wave)
//          + prefetch of gathered x1 rows.
// Phase 2: W_edge(16x384) = P(16x32) @ tp_weight(32x384) via WMMA (24 tiles/8 waves).
// Phase 3: per-edge tensor product + atomic scatter into msg.
#define TPB 4     // 16-edge tiles per block -> 64 edges per block
__global__ void edge_kernel(const float* __restrict__ x1,
                            const float* __restrict__ edge_query,
                            const float* __restrict__ edge_sh,
                            const float* __restrict__ tp_keys,
                            const float* __restrict__ tp_weight,
                            const int*   __restrict__ edge_src,
                            const int*   __restrict__ edge_dst,
                            float* __restrict__ msg) {
  __shared__ float s_keys[NUM_FILTERS * DIM_KEY];   //  8 KB (filter-major)
  __shared__ float s_q[16][DIM_KEY];                //  4 KB
  __shared__ float s_p[16][NUM_FILTERS];            //  2 KB
  __shared__ float s_wedge[16][W_NUMEL];            // 24 KB

  const int tid = threadIdx.x;
  for (int i = tid; i < NUM_FILTERS * DIM_KEY; i += 256) s_keys[i] = tp_keys[i];

  const int wv = tid >> 5, lane = tid & 31;
  const int hf = lane >> 4, lq = lane & 15;
  const long blk0 = (long)blockIdx.x * (TPB * 16);

  for (int tile = 0; tile < TPB; ++tile) {
    const long tbase = blk0 + (long)tile * 16;

    // ---- stage queries + prefetch source rows (2 edges per wave) ----
    #pragma unroll
    for (int j = 0; j < 2; ++j) {
      const int  l = wv + 8 * j;
      const long e = tbase + l;
      const float* q = edge_query + e * DIM_KEY;
      s_q[l][lane]      = q[lane];
      s_q[l][lane + 32] = q[lane + 32];
      const int src = edge_src[e];
      const float* xrow = x1 + (long)src * D_IN;
      if (lane < 10)                        // 10 x 128B covers 1280B row
        __builtin_prefetch(xrow + lane * 32, 0, 3);
    }
    __syncthreads();

    // ---- phase 1: logits + softmax (lane == filter) ----
    #pragma unroll
    for (int j = 0; j < 2; ++j) {
      const int l = wv + 8 * j;
      float lg = 0.f;
      #pragma unroll
      for (int k = 0; k < DIM_KEY; ++k)
        lg = fmaf(s_q[l][k], s_keys[lane * DIM_KEY + k], lg);
      lg *= 0.125f;                         // 1/sqrt(DIM_KEY)
      float mx = lg;
      #pragma unroll
      for (int off = 16; off > 0; off >>= 1) mx = fmaxf(mx, __shfl_xor(mx, off, 32));
      const float ex = __expf(lg - mx);
      float sm = ex;
      #pragma unroll
      for (int off = 16; off > 0; off >>= 1) sm += __shfl_xor(sm, off, 32);
      s_p[l][lane] = ex / sm;
    }
    __syncthreads();

    // ---- phase 2: W_edge = P @ tp_weight via WMMA (24 col tiles / 8 waves) ----
    #pragma unroll
    for (int tj = 0; tj < 3; ++tj) {
      const int nt  = wv + 8 * tj;          // column tile (16 cols)
      const int col = nt * 16 + lq;
      v8f acc = {0.f,0.f,0.f,0.f,0.f,0.f,0.f,0.f};
      #pragma unroll
      for (int k0 = 0; k0 < NUM_FILTERS; k0 += 4) {
        const int ka = k0 + 2 * hf;
        v2f a = *(const v2f*)&s_p[lq][ka];            // b64 LDS read
        v2f b;
        b.x = tp_weight[(long)ka       * W_NUMEL + col];
        b.y = tp_weight[(long)(ka + 1) * W_NUMEL + col];
        acc = wmma4(a, b, acc);
      }
      #pragma unroll
      for (int r = 0; r < 8; ++r)
        s_wedge[r + 8 * hf][col] = acc[r];
    }
    __syncthreads();

    // ---- phase 3: tensor product + atomic scatter (2 edges per wave) ----
    #pragma unroll
    for (int j = 0; j < 2; ++j) {
      const int  l = wv + 8 * j;
      const long e = tbase + l;
      const int src = edge_src[e];
      const int dst = edge_dst[e];
      const float* xr  = x1 + (long)src * D_IN;
      float*       mrw = msg + (long)dst * MSG_W;
      const float sh0  = edge_sh[e * 4 + 0];
      const float sh1x = edge_sh[e * 4 + 1];
      const float sh1y = edge_sh[e * 4 + 2];
      const float sh1z = edge_sh[e * 4 + 3];

      float w[12];
      #pragma unroll
      for (int t = 0; t < 12; ++t) w[t] = s_wedge[l][lane + 32 * t];

      // o0, o1 (scalar path)
      #pragma unroll
      for (int t = 0; t < 4; ++t) {
        const int u  = lane + 32 * t;
        const float xs = xr[u];
        atomicAddF(&mrw[u], w[t] * xs * sh0);
        const float b = w[4 + t] * xs;
        atomicAddF(&mrw[MUL_S + 3 * u + 0], b * sh1x);
        atomicAddF(&mrw[MUL_S + 3 * u + 1], b * sh1y);
        atomicAddF(&mrw[MUL_S + 3 * u + 2], b * sh1z);
      }
      // o2, o3 (vector path)
      #pragma unroll
      for (int t = 0; t < 2; ++t) {
        const int v = lane + 32 * t;
        const float vx = xr[MUL_S + 3 * v + 0];
        const float vy = xr[MUL_S + 3 * v + 1];
        const float vz = xr[MUL_S + 3 * v + 2];
        const float c = w[8 + t] * sh0;
        atomicAddF(&mrw[512 + 3 * v + 0], c * vx);
        atomicAddF(&mrw[512 + 3 * v + 1], c * vy);
        atomicAddF(&mrw[512 + 3 * v + 2], c * vz);
        const float dv = vx * sh1x + vy * sh1y + vz * sh1z;
        atomicAddF(&mrw[704 + v], w[10 + t] * dv * INV_SQRT3);
      }
    }
    __syncthreads();   // protect s_q/s_p/s_wedge before next tile
  }
}

// --------------------------------------- kernel C: output linears (WMMA) + add
__global__ void node_out_kernel(const float* __restrict__ msg,
                                const float* __restrict__ W2_s0,
                                const float* __restrict__ W2_s3,
                                const float* __restrict__ W2_v1,
                                const float* __restrict__ W2_v2,
                                float* __restrict__ out) {
  const int node0 = blockIdx.x * 16;
  const int wv   = threadIdx.x >> 5;
  const int lane = threadIdx.x & 31;
  const int hf   = lane >> 4;
  const int lq   = lane & 15;

  for (int task = wv; task < 20; task += 8) {
    v8f acc = {0.f,0.f,0.f,0.f,0.f,0.f,0.f,0.f};
    const float* mrow = msg + (long)(node0 + lq) * MSG_W;
    if (task < 8) {
      const int nt = task;
      #pragma unroll
      for (int k0 = 0; k0 < MUL_S; k0 += 4) {       // m0 @ W2_s0
        const int ka = k0 + 2 * hf;
        v2f a = *(const v2f*)(mrow + ka);
        v2f b;
        b.x = W2_s0[(long)ka       * MUL_S + nt * 16 + lq];
        b.y = W2_s0[(long)(ka + 1) * MUL_S + nt * 16 + lq];
        acc = wmma4(a, b, acc);
      }
      #pragma unroll
      for (int k0 = 0; k0 < MUL_V; k0 += 4) {       // m3 @ W2_s3
        const int ka = k0 + 2 * hf;
        v2f a = *(const v2f*)(mrow + 704 + ka);
        v2f b;
        b.x = W2_s3[(long)ka       * MUL_S + nt * 16 + lq];
        b.y = W2_s3[(long)(ka + 1) * MUL_S + nt * 16 + lq];
        acc = wmma4(a, b, acc);
      }
      #pragma unroll
      for (int r = 0; r < 8; ++r) {
        const int mr = r + 8 * hf;
        const long idx = (long)(node0 + mr) * D_IN + nt * 16 + lq;
        out[idx] = out[idx] + acc[r] * OUT_SCALE;
      }
    } else {
      const int t2 = task - 8;        // 0..11
      const int ci = t2 >> 2;
      const int nt = t2 & 3;
      #pragma unroll
      for (int k0 = 0; k0 < MUL_S; k0 += 4) {       // m1[:,:,ci] @ W2_v1
        const int ka = k0 + 2 * hf;
        v2f a, b;
        a.x = mrow[MUL_S + 3 * ka + ci];
        a.y = mrow[MUL_S + 3 * (ka + 1) + ci];
        b.x = W2_v1[(long)ka       * MUL_V + nt * 16 + lq];
        b.y = W2_v1[(long)(ka + 1) * MUL_V + nt * 16 + lq];
        acc = wmma4(a, b, acc);
      }
      #pragma unroll
      for (int k0 = 0; k0 < MUL_V; k0 += 4) {       // m2[:,:,ci] @ W2_v2
        const int ka = k0 + 2 * hf;
        v2f a, b;
        a.x = mrow[512 + 3 * ka + ci];
        a.y = mrow[512 + 3 * (ka + 1) + ci];
        b.x = W2_v2[(long)ka       * MUL_V + nt * 16 + lq];
        b.y = W2_v2[(long)(ka + 1) * MUL_V + nt * 16 + lq];
        acc = wmma4(a, b, acc);
      }
      #pragma unroll
      for (int r = 0; r < 8; ++r) {
        const int mr = r + 8 * hf;
        const long idx = (long)(node0 + mr) * D_IN + MUL_S + 3 * (nt * 16 + lq) + ci;
        out[idx] = out[idx] + acc[r] * OUT_SCALE;
      }
    }
  }
}

extern "C" void kernel_launch(void* const* d_in, const int* in_sizes, int n_in,
                              void* d_out, int out_size, void* d_ws, size_t ws_size,
                              hipStream_t stream) {
  const float* x          = (const float*)d_in[0];
  const float* edge_query = (const float*)d_in[1];
  const float* edge_sh    = (const float*)d_in[2];
  const float* W_si_s     = (const float*)d_in[3];
  const float* W_si_v     = (const float*)d_in[4];
  const float* W_l1_s     = (const float*)d_in[5];
  const float* W_l1_v     = (const float*)d_in[6];
  const float* tp_keys    = (const float*)d_in[7];
  const float* tp_weight  = (const float*)d_in[8];
  const float* W2_s0      = (const float*)d_in[9];
  const float* W2_s3      = (const float*)d_in[10];
  const float* W2_v1      = (const float*)d_in[11];
  const float* W2_v2      = (const float*)d_in[12];
  const int*   edge_src   = (const int*)d_in[13];
  const int*   edge_dst   = (const int*)d_in[14];
  float* out = (float*)d_out;

  float* x1  = (float*)d_ws;                       // 10000*320 f32 = 12.8 MB
  float* msg = x1 + (long)N_NODES * D_IN;          // 10000*768 f32 = 30.7 MB

  const long msg_elems = (long)N_NODES * MSG_W;
  zero_kernel<<<4096, 256, 0, stream>>>(msg, msg_elems);

  node_linear_kernel<<<N_NODES / 16, 256, 0, stream>>>(
      x, W_si_s, W_si_v, W_l1_s, W_l1_v, out, x1);

  edge_kernel<<<N_EDGES / (TPB * 16), 256, 0, stream>>>(
      x1, edge_query, edge_sh, tp_keys, tp_weight, edge_src, edge_dst, msg);

  node_out_kernel<<<N_NODES / 16, 256, 0, stream>>>(
      msg, W2_s0, W2_s3, W2_v1, W2_v2, out);
}